// PointNetEncoder_455266533580
// MI455X (gfx1250) — compile-verified
//
#include <hip/hip_runtime.h>
#include <hip/hip_bf16.h>

typedef __attribute__((ext_vector_type(2))) float v2f;
typedef __attribute__((ext_vector_type(8))) float v8f;

#define BATCH  16
#define NPTS   4096
#define NCATS  8
#define HEADF  4
#define SHAPE  64
#define CIN    68      // HEAD + ENC_OUT
#define ENCH   128
#define ENCO   64
#define H1D    128
#define H2D    256
#define LATENT 512

// 32 points (two 16-row WMMA tiles) per wave.
// LDS row strides chosen so (stride % 64) in {2,4} -> 16 rows hit 16 distinct
// banks for column (same-k) reads, and per-lane b64 pairs don't collide.
#define LD_IN  68    // tileIn : 32 x 68 (head+code)
#define LD_E   66    // encBuf : 32 x 64 expert outputs
#define LD_A   130   // bufA   : 32 x 128
#define LD_C   258   // bufBC  : 32 x 256

#define OFF_ENC  (32 * LD_IN)
#define OFF_A    (OFF_ENC + 32 * LD_E)
#define OFF_BC   (OFF_A + 32 * LD_A)
#define OFF_CATS (OFF_BC + 32 * LD_C)
#define SMEM_FLOATS (OFF_CATS + 32)
#define SMEM_BYTES  (SMEM_FLOATS * 4)     // 66944 B < 320 KB/WGP

// One B fetch feeds both 16-row A tiles: 1 global load-pair per 2 WMMAs.
// A layout (32-bit 16x4): lanes 0-15 hold M=lane, k=k0,k0+1; lanes 16-31 hold
// M=lane-16, k=k0+2,k0+3. B mirrors with N=lane&15.
template<bool KN>
__device__ __forceinline__ void wmma_tile2(const float* act, int ldA,   // 32 rows in LDS
                                           const float* __restrict__ W, int ldW,
                                           int n0, int K, int lane,
                                           v8f& acc0, v8f& acc1) {
  const int half = lane >> 4;
  const int l16  = lane & 15;
  for (int k0 = 0; k0 < K; k0 += 4) {
    const int ka = k0 + 2 * half;
    v2f bb;
    if (KN) { bb.x = W[(size_t)ka * ldW + n0 + l16];
              bb.y = W[(size_t)(ka + 1) * ldW + n0 + l16]; }
    else    { bb.x = W[(size_t)(n0 + l16) * ldW + ka];
              bb.y = W[(size_t)(n0 + l16) * ldW + ka + 1]; }
    v2f a0, a1;
    a0.x = act[l16 * ldA + ka];        a0.y = act[l16 * ldA + ka + 1];
    a1.x = act[(16 + l16) * ldA + ka]; a1.y = act[(16 + l16) * ldA + ka + 1];
    acc0 = __builtin_amdgcn_wmma_f32_16x16x4_f32(false, a0, false, bb,
                                                 (short)0, acc0, false, false);
    acc1 = __builtin_amdgcn_wmma_f32_16x16x4_f32(false, a1, false, bb,
                                                 (short)0, acc1, false, false);
  }
}

__global__ void pn_init_out(float* __restrict__ out, int n) {
  int i = blockIdx.x * blockDim.x + threadIdx.x;
  if (i < n) out[i] = __int_as_float(0xff800000); // -inf
}

__global__ __launch_bounds__(32)
void pn_moe_kernel(const float* __restrict__ x,  const int* __restrict__ cats,
                   const float* __restrict__ eW1, const float* __restrict__ eb1,
                   const float* __restrict__ eW2, const float* __restrict__ eb2,
                   const float* __restrict__ eW3, const float* __restrict__ eb3,
                   const float* __restrict__ cW1, const float* __restrict__ cb1,
                   const float* __restrict__ cW2, const float* __restrict__ cb2,
                   const float* __restrict__ cW3, const float* __restrict__ cb3,
                   float* __restrict__ out) {
  extern __shared__ float smem[];
  float* tileIn = smem;                 // 32 x 68  head+code (points x features)
  float* encB   = smem + OFF_ENC;       // 32 x 64  selected expert outputs
  float* bufA   = smem + OFF_A;         // 32 x 128 activations
  float* bufBC  = smem + OFF_BC;        // 32 x 128/256 activations
  int*   catsL  = (int*)(smem + OFF_CATS);

  const int lane = threadIdx.x;
  const int half = lane >> 4;
  const int l16  = lane & 15;
  const int tile = blockIdx.x;          // one wave == 32 points
  const int b    = tile >> 7;           // P/32 = 128 tiles per batch
  const int p0   = (tile & 127) << 5;

  // ---- stage 32x68 input tile from x (B, 68, P): coalesced 128B rows ----
  const float* xb = x + (size_t)b * CIN * NPTS;
  for (int idx = lane; idx < CIN * 32; idx += 32) {
    int f = idx >> 5, i = idx & 31;
    tileIn[i * LD_IN + f] = xb[(size_t)f * NPTS + p0 + i];
  }
  catsL[lane] = cats[b * NPTS + p0 + lane];
  __syncthreads();

  // wave-uniform category presence mask
  unsigned pmask = 0;
  #pragma unroll
  for (int i = 0; i < 32; ++i) pmask |= 1u << catsL[i];

  // ---- mixture-of-experts encoder (only categories present in tile) ----
  for (int c = 0; c < NCATS; ++c) {
    if (!((pmask >> c) & 1u)) continue;
    const float* W1 = eW1 + (size_t)c * SHAPE * ENCH;
    const float* W2 = eW2 + (size_t)c * ENCH * ENCH;
    const float* W3 = eW3 + (size_t)c * ENCH * ENCO;
    const float* b1 = eb1 + c * ENCH;
    const float* b2 = eb2 + c * ENCH;
    const float* b3 = eb3 + c * ENCO;

    // L1: code(32x64) @ W1(64x128) + b1, relu -> bufA
    for (int n0 = 0; n0 < ENCH; n0 += 16) {
      v8f a0 = {}, a1 = {};
      wmma_tile2<true>(tileIn + HEADF, LD_IN, W1, ENCH, n0, SHAPE, lane, a0, a1);
      float bv = b1[n0 + l16];
      #pragma unroll
      for (int j = 0; j < 8; ++j) {
        int r = j + 8 * half;
        bufA[r * LD_A + n0 + l16]        = fmaxf(a0[j] + bv, 0.f);
        bufA[(16 + r) * LD_A + n0 + l16] = fmaxf(a1[j] + bv, 0.f);
      }
    }
    __syncthreads();

    // L2: (32x128) @ W2(128x128) + b2, relu -> bufBC
    for (int n0 = 0; n0 < ENCH; n0 += 16) {
      v8f a0 = {}, a1 = {};
      wmma_tile2<true>(bufA, LD_A, W2, ENCH, n0, ENCH, lane, a0, a1);
      float bv = b2[n0 + l16];
      #pragma unroll
      for (int j = 0; j < 8; ++j) {
        int r = j + 8 * half;
        bufBC[r * LD_C + n0 + l16]        = fmaxf(a0[j] + bv, 0.f);
        bufBC[(16 + r) * LD_C + n0 + l16] = fmaxf(a1[j] + bv, 0.f);
      }
    }
    __syncthreads();

    // L3: (32x128) @ W3(128x64) + b3 -> select rows with cats==c into encB
    for (int n0 = 0; n0 < ENCO; n0 += 16) {
      v8f a0 = {}, a1 = {};
      wmma_tile2<true>(bufBC, LD_C, W3, ENCO, n0, ENCH, lane, a0, a1);
      float bv = b3[n0 + l16];
      #pragma unroll
      for (int j = 0; j < 8; ++j) {
        int r = j + 8 * half;
        if (catsL[r] == c)      encB[r * LD_E + n0 + l16]        = a0[j] + bv;
        if (catsL[16 + r] == c) encB[(16 + r) * LD_E + n0 + l16] = a1[j] + bv;
      }
    }
    __syncthreads();
  }

  // ---- shared classifier MLP ----
  // c1: xi = [head(4) | enc(64)] @ cW1(128,68)^T + cb1, relu -> bufA
  //     k-step 0 reads head straight from tileIn; k-steps 1..16 read encB.
  for (int n0 = 0; n0 < H1D; n0 += 16) {
    v8f a0 = {}, a1 = {};
    {   // k = 0..3 : head features
      const int ka = 2 * half;
      v2f bb, v0, v1;
      bb.x = cW1[(size_t)(n0 + l16) * CIN + ka];
      bb.y = cW1[(size_t)(n0 + l16) * CIN + ka + 1];
      v0.x = tileIn[l16 * LD_IN + ka];        v0.y = tileIn[l16 * LD_IN + ka + 1];
      v1.x = tileIn[(16 + l16) * LD_IN + ka]; v1.y = tileIn[(16 + l16) * LD_IN + ka + 1];
      a0 = __builtin_amdgcn_wmma_f32_16x16x4_f32(false, v0, false, bb, (short)0, a0, false, false);
      a1 = __builtin_amdgcn_wmma_f32_16x16x4_f32(false, v1, false, bb, (short)0, a1, false, false);
    }
    for (int k0 = 4; k0 < CIN; k0 += 4) {   // k = 4..67 : expert features
      const int ka = k0 + 2 * half;
      v2f bb, v0, v1;
      bb.x = cW1[(size_t)(n0 + l16) * CIN + ka];
      bb.y = cW1[(size_t)(n0 + l16) * CIN + ka + 1];
      v0.x = encB[l16 * LD_E + ka - 4];        v0.y = encB[l16 * LD_E + ka - 3];
      v1.x = encB[(16 + l16) * LD_E + ka - 4]; v1.y = encB[(16 + l16) * LD_E + ka - 3];
      a0 = __builtin_amdgcn_wmma_f32_16x16x4_f32(false, v0, false, bb, (short)0, a0, false, false);
      a1 = __builtin_amdgcn_wmma_f32_16x16x4_f32(false, v1, false, bb, (short)0, a1, false, false);
    }
    float bv = cb1[n0 + l16];
    #pragma unroll
    for (int j = 0; j < 8; ++j) {
      int r = j + 8 * half;
      bufA[r * LD_A + n0 + l16]        = fmaxf(a0[j] + bv, 0.f);
      bufA[(16 + r) * LD_A + n0 + l16] = fmaxf(a1[j] + bv, 0.f);
    }
  }
  __syncthreads();

  // c2: (32x128) @ cW2(256,128)^T + cb2, relu -> bufBC (32x256)
  for (int n0 = 0; n0 < H2D; n0 += 16) {
    v8f a0 = {}, a1 = {};
    wmma_tile2<false>(bufA, LD_A, cW2, H1D, n0, H1D, lane, a0, a1);
    float bv = cb2[n0 + l16];
    #pragma unroll
    for (int j = 0; j < 8; ++j) {
      int r = j + 8 * half;
      bufBC[r * LD_C + n0 + l16]        = fmaxf(a0[j] + bv, 0.f);
      bufBC[(16 + r) * LD_C + n0 + l16] = fmaxf(a1[j] + bv, 0.f);
    }
  }
  __syncthreads();

  // c3: (32x256) @ cW3(512,256)^T + cb3 -> max over 32 points -> atomic fmax
  float* ob = out + b * LATENT;
  for (int n0 = 0; n0 < LATENT; n0 += 16) {
    v8f a0 = {}, a1 = {};
    wmma_tile2<false>(bufBC, LD_C, cW3, H2D, n0, H2D, lane, a0, a1);
    float m = fmaxf(a0[0], a1[0]);
    #pragma unroll
    for (int j = 1; j < 8; ++j) m = fmaxf(m, fmaxf(a0[j], a1[j]));
    m += cb3[n0 + l16];                   // bias is per-N, add after row-max
    m = fmaxf(m, __shfl_xor(m, 16, 32));  // fold the two 8-row halves
    if (lane < 16)
      __hip_atomic_fetch_max(&ob[n0 + l16], m, __ATOMIC_RELAXED,
                             __HIP_MEMORY_SCOPE_AGENT);
  }
}

extern "C" void kernel_launch(void* const* d_in, const int* in_sizes, int n_in,
                              void* d_out, int out_size, void* d_ws, size_t ws_size,
                              hipStream_t stream) {
  (void)in_sizes; (void)n_in; (void)d_ws; (void)ws_size;
  const float* x   = (const float*)d_in[0];
  const int*  cats = (const int*)  d_in[1];
  const float* eW1 = (const float*)d_in[2];
  const float* eb1 = (const float*)d_in[3];
  const float* eW2 = (const float*)d_in[4];
  const float* eb2 = (const float*)d_in[5];
  const float* eW3 = (const float*)d_in[6];
  const float* eb3 = (const float*)d_in[7];
  const float* cW1 = (const float*)d_in[8];
  const float* cb1 = (const float*)d_in[9];
  const float* cW2 = (const float*)d_in[10];
  const float* cb2 = (const float*)d_in[11];
  const float* cW3 = (const float*)d_in[12];
  const float* cb3 = (const float*)d_in[13];
  float* out = (float*)d_out;

  // allow >64KB dynamic LDS (gfx1250 WGP has 320KB); idempotent, capture-safe
  (void)hipFuncSetAttribute((const void*)pn_moe_kernel,
                            hipFuncAttributeMaxDynamicSharedMemorySize,
                            SMEM_BYTES);

  pn_init_out<<<(out_size + 255) / 256, 256, 0, stream>>>(out, out_size);

  const int ntiles = (BATCH * NPTS) / 32;   // 2048 waves, 32 points each
  pn_moe_kernel<<<ntiles, 32, SMEM_BYTES, stream>>>(
      x, cats, eW1, eb1, eW2, eb2, eW3, eb3,
      cW1, cb1, cW2, cb2, cW3, cb3, out);
}